// STAR_6828998001587
// MI455X (gfx1250) — compile-verified
//
#include <hip/hip_runtime.h>
#include <math.h>

// Problem constants (match reference)
static constexpr int NB = 16;      // batch
static constexpr int NS = 2048;    // seq
static constexpr int ND = 1024;    // model dim
static constexpr int NH = 8;       // heads
static constexpr int HD = 128;     // head dim
static constexpr int NE = 8;       // experts / strategies
static constexpr int GD = 128;     // gate dim
static constexpr int FF = 4096;    // ffn dim
static constexpr int PCH = 32;     // pooling chunks
static constexpr int JCH = 8;      // j chunks for weighted sum

typedef __attribute__((ext_vector_type(2))) float v2f;
typedef __attribute__((ext_vector_type(4))) float v4f;
typedef __attribute__((ext_vector_type(8))) float v8f;

// ---------------------------------------------------------------------------
// Generic fp32 WMMA tile: C[16,16] (+bias, *cscale) = A[16,K] * B[K, ntile*16..+16]
// A-layout (16x4 f32, ISA 7.12.2): lane m=lane&15; VGPR0 holds K=(lane>=16?2:0), VGPR1 K+1
// B-layout (4x16 f32): lane n=lane&15; VGPR0 holds K=(lane>=16?2:0) row, VGPR1 K+1
// C/D-layout (16x16 f32): VGPR i -> M = i + (lane>=16 ? 8 : 0), N = lane&15
// BT=true : B[k,n] stored as Bm[n*ldb + k] (i.e. weights used as W.T)
// BT=false: B[k,n] stored as Bm[k*ldb + n]
// ---------------------------------------------------------------------------
template <bool BT>
__device__ inline void wmma_gemm_tile(const float* __restrict__ A, int lda,
                                      const float* __restrict__ Bm, int ldb,
                                      const float* __restrict__ bias,
                                      float* __restrict__ C, int ldc,
                                      int K, int ntile, float cscale) {
  const int lane = threadIdx.x & 31;
  const int half = lane >> 4;      // 0 or 1 -> K offset 0 or 2
  const int mr   = lane & 15;      // M row of A / N col of B
  const int ncol = ntile * 16 + mr;
  v8f c = {};
  for (int k0 = 0; k0 < K; k0 += 4) {
    const int ka = k0 + half * 2;
    v2f a, b;
    a.x = A[mr * lda + ka];
    a.y = A[mr * lda + ka + 1];
    if (BT) {
      b.x = Bm[(size_t)ncol * ldb + ka];
      b.y = Bm[(size_t)ncol * ldb + ka + 1];
    } else {
      b.x = Bm[(size_t)ka * ldb + ncol];
      b.y = Bm[(size_t)(ka + 1) * ldb + ncol];
    }
    c = __builtin_amdgcn_wmma_f32_16x16x4_f32(false, a, false, b, (short)0, c,
                                              false, false);
  }
  const float bvv = bias ? bias[ncol] : 0.0f;
  const int mbase = half * 8;
#pragma unroll
  for (int i = 0; i < 8; ++i)
    C[(size_t)(mbase + i) * ldc + ncol] = c[i] * cscale + bvv;
}

// --- 1/2: column mean of hidden_states, two-stage (deterministic) -----------
__global__ void k_pool_partial(const float* __restrict__ x, float* __restrict__ partial) {
  const int bp = blockIdx.x;                 // 0..NB*PCH-1
  const int b = bp / PCH, pc = bp % PCH;
  const int d = blockIdx.y * 256 + threadIdx.x;
  const float* xr = x + ((size_t)b * NS + (size_t)pc * (NS / PCH)) * ND + d;
  float acc = 0.f;
  for (int s = 0; s < NS / PCH; ++s) acc += xr[(size_t)s * ND];
  partial[(size_t)bp * ND + d] = acc;
}

__global__ void k_pool_reduce(const float* __restrict__ partial, float* __restrict__ pooled) {
  const int b = blockIdx.x;
  const int d = blockIdx.y * 256 + threadIdx.x;
  float acc = 0.f;
  for (int pc = 0; pc < PCH; ++pc) acc += partial[((size_t)b * PCH + pc) * ND + d];
  pooled[(size_t)b * ND + d] = acc * (1.0f / NS);
}

// --- 3: q = pooled @ Wq.T + bq  (and reused for attn_out = ctx @ Wc.T + bc) -
__global__ void k_gemm16_bt(const float* __restrict__ A, const float* __restrict__ W,
                            const float* __restrict__ bias, float* __restrict__ C) {
  const int tile = blockIdx.x * 8 + (threadIdx.x >> 5);  // 0..63 over N=1024
  wmma_gemm_tile<true>(A, ND, W, ND, bias, C, ND, ND, tile, 1.0f);
}

// --- 4: sbias[b,h] = scale * q[b,h]·bk_h ------------------------------------
__global__ void k_sbias(const float* __restrict__ q, const float* __restrict__ bk,
                        float* __restrict__ sbias, float scale) {
  const int t = threadIdx.x;  // 0..127 = b*8+h
  const int b = t >> 3, h = t & 7;
  float acc = 0.f;
  for (int dd = 0; dd < HD; ++dd) acc += q[(size_t)b * ND + h * HD + dd] * bk[h * HD + dd];
  sbias[t] = acc * scale;
}

// --- 5: t[b,h,c] = scale * sum_dd q[b,h*HD+dd] * Wk[h*HD+dd, c] -------------
__global__ void k_gemm_th(const float* __restrict__ q, const float* __restrict__ Wk,
                          float* __restrict__ th, float scale) {
  const int tile = blockIdx.x * 8 + (threadIdx.x >> 5);  // 0..511
  const int h = tile >> 6, nt = tile & 63;
  wmma_gemm_tile<false>(q + h * HD, ND, Wk + (size_t)h * HD * ND, ND, nullptr,
                        th + (size_t)h * ND, NH * ND, HD, nt, scale);
}

// --- 6: scores[b,h,j] = x_cat[b,j]·t[b,h] + sbias[b,h]  (warp per (b,j)) ----
__global__ void k_scores(const float* __restrict__ x, const float* __restrict__ pooled,
                         const float* __restrict__ th, const float* __restrict__ sbias,
                         float* __restrict__ scores) {
  const int warp = blockIdx.x * 8 + (threadIdx.x >> 5);
  const int lane = threadIdx.x & 31;
  if (warp >= NB * (NS + 1)) return;
  const int b = warp / (NS + 1);
  const int j = warp % (NS + 1);
  const float* xr = (j == 0) ? (pooled + (size_t)b * ND)
                             : (x + ((size_t)b * NS + (j - 1)) * ND);
  const float* tb = th + (size_t)b * NH * ND;
  float acc[NH];
#pragma unroll
  for (int h = 0; h < NH; ++h) acc[h] = 0.f;
  for (int c = lane; c < ND; c += 32) {
    const float xv = xr[c];
#pragma unroll
    for (int h = 0; h < NH; ++h) acc[h] += xv * tb[(size_t)h * ND + c];
  }
#pragma unroll
  for (int h = 0; h < NH; ++h)
    for (int off = 16; off; off >>= 1) acc[h] += __shfl_xor(acc[h], off, 32);
  if (lane == 0) {
#pragma unroll
    for (int h = 0; h < NH; ++h)
      scores[((size_t)b * NH + h) * (NS + 1) + j] = acc[h] + sbias[b * NH + h];
  }
}

// --- 7: softmax over j per (b,h), in place ----------------------------------
__global__ void k_softmax(float* __restrict__ scores) {
  float* row = scores + (size_t)blockIdx.x * (NS + 1);
  __shared__ float red[256];
  const int t = threadIdx.x;
  float m = -3.4e38f;
  for (int j = t; j < NS + 1; j += 256) m = fmaxf(m, row[j]);
  red[t] = m; __syncthreads();
  for (int s = 128; s; s >>= 1) { if (t < s) red[t] = fmaxf(red[t], red[t + s]); __syncthreads(); }
  m = red[0]; __syncthreads();
  float sum = 0.f;
  for (int j = t; j < NS + 1; j += 256) { float e = expf(row[j] - m); row[j] = e; sum += e; }
  red[t] = sum; __syncthreads();
  for (int s = 128; s; s >>= 1) { if (t < s) red[t] += red[t + s]; __syncthreads(); }
  const float inv = 1.0f / red[0];
  for (int j = t; j < NS + 1; j += 256) row[j] *= inv;
}

// --- 8/9: u[b,h,c] = sum_j attn[b,h,j] * x_cat[b,j,c], two-stage over j -----
__global__ void k_upart(const float* __restrict__ x, const float* __restrict__ pooled,
                        const float* __restrict__ attn, float* __restrict__ upart) {
  const int b = blockIdx.y, jc = blockIdx.z;
  const int c = blockIdx.x * 256 + threadIdx.x;
  const int jstart = jc * 256;
  const int jend = (jc == JCH - 1) ? (NS + 1) : jstart + 256;
  const int jlen = jend - jstart;
  __shared__ float sa[NH][257];
  for (int idx = threadIdx.x; idx < NH * jlen; idx += 256) {
    const int h = idx / jlen, jj = idx % jlen;
    sa[h][jj] = attn[((size_t)b * NH + h) * (NS + 1) + jstart + jj];
  }
  __syncthreads();
  float acc[NH] = {};
  for (int j = jstart; j < jend; ++j) {
    const float* xr = (j == 0) ? (pooled + (size_t)b * ND)
                               : (x + ((size_t)b * NS + (j - 1)) * ND);
    const float xv = xr[c];
    const int jj = j - jstart;
#pragma unroll
    for (int h = 0; h < NH; ++h) acc[h] += sa[h][jj] * xv;
  }
#pragma unroll
  for (int h = 0; h < NH; ++h)
    upart[(((size_t)jc * NB + b) * NH + h) * ND + c] = acc[h];
}

__global__ void k_ureduce(const float* __restrict__ upart, float* __restrict__ u) {
  const size_t idx = (size_t)blockIdx.x * 256 + threadIdx.x;  // over NB*NH*ND
  float acc = 0.f;
  for (int jc = 0; jc < JCH; ++jc) acc += upart[(size_t)jc * NB * NH * ND + idx];
  u[idx] = acc;
}

// --- 10: ctx[b, h*HD+n] = Wv_h @ u[b,h] + bv_h  (WMMA, per head) ------------
__global__ void k_gemm_ctx(const float* __restrict__ u, const float* __restrict__ Wv,
                           const float* __restrict__ bv, float* __restrict__ ctx) {
  const int tile = blockIdx.x * 8 + (threadIdx.x >> 5);  // 0..63
  const int h = tile >> 3, nt = tile & 7;
  wmma_gemm_tile<true>(u + (size_t)h * ND, NH * ND, Wv + (size_t)h * HD * ND, ND,
                       bv + h * HD, ctx + h * HD, ND, ND, nt, 1.0f);
}

// --- 12: gate (strategy-routed), also writes gate into d_out tail -----------
__global__ void k_gate(const float* __restrict__ attn_out, const int* __restrict__ strategy,
                       const float* __restrict__ Wgin, const float* __restrict__ bgin,
                       const float* __restrict__ Wgout, float* __restrict__ gate,
                       float* __restrict__ out_gate) {
  const int b = blockIdx.x, t = threadIdx.x;  // t = gd, 0..127
  const int e = strategy[b];
  const float* w = Wgin + ((size_t)e * GD + t) * ND;
  const float* a = attn_out + (size_t)b * ND;
  float acc = 0.f;
  for (int d = 0; d < ND; ++d) acc += a[d] * w[d];
  const float g = fmaxf(acc + bgin[e * GD + t], 0.f);
  __shared__ float red[128];
  red[t] = g * Wgout[e * GD + t]; __syncthreads();
  for (int s = 64; s; s >>= 1) { if (t < s) red[t] += red[t + s]; __syncthreads(); }
  if (t == 0) {
    const float gt = 1.f / (1.f + expf(-red[0]));
    gate[b] = gt;
    out_gate[b] = gt;
  }
}

// --- 13: h1 = gelu(attn_out @ Wfc1[e].T + bfc1[e])  (warp per output) -------
// Weight rows are single-use streams: non-temporal loads keep hidden_states
// resident in the 192MB L2 for the later blend pass.
__global__ void k_fc1(const float* __restrict__ attn_out, const int* __restrict__ strategy,
                      const float* __restrict__ Wfc1, const float* __restrict__ bfc1,
                      float* __restrict__ h1) {
  const int warp = blockIdx.x * 8 + (threadIdx.x >> 5);  // 0..NB*FF-1
  const int lane = threadIdx.x & 31;
  const int b = warp / FF, f = warp % FF;
  const int e = strategy[b];
  const float* w = Wfc1 + ((size_t)e * FF + f) * ND;
  const float* a = attn_out + (size_t)b * ND;
  float acc = 0.f;
  for (int d = lane; d < ND; d += 32)
    acc += a[d] * __builtin_nontemporal_load(&w[d]);
  for (int off = 16; off; off >>= 1) acc += __shfl_xor(acc, off, 32);
  if (lane == 0) {
    const float z = acc + bfc1[(size_t)e * FF + f];
    h1[(size_t)b * FF + f] = 0.5f * z * (1.f + erff(z * 0.70710678118654752f));
  }
}

// --- 14: proj = h1 @ Wfc2[e].T + bfc2[e]  (warp per output) -----------------
__global__ void k_fc2(const float* __restrict__ h1, const int* __restrict__ strategy,
                      const float* __restrict__ Wfc2, const float* __restrict__ bfc2,
                      float* __restrict__ proj) {
  const int warp = blockIdx.x * 8 + (threadIdx.x >> 5);  // 0..NB*ND-1
  const int lane = threadIdx.x & 31;
  const int b = warp / ND, d = warp % ND;
  const int e = strategy[b];
  const float* w = Wfc2 + ((size_t)e * ND + d) * FF;
  const float* hh = h1 + (size_t)b * FF;
  float acc = 0.f;
  for (int f = lane; f < FF; f += 32)
    acc += hh[f] * __builtin_nontemporal_load(&w[f]);
  for (int off = 16; off; off >>= 1) acc += __shfl_xor(acc, off, 32);
  if (lane == 0) proj[(size_t)b * ND + d] = acc + bfc2[(size_t)e * ND + d];
}

// --- 15: out = gate*proj + (1-gate)*x  (streaming, v4, NT hints) ------------
// x reads here are last-use; out is write-once never-reread: both non-temporal
// so the 128MiB output stream does not rinse through L2.
__global__ void k_blend(const float* __restrict__ x, const float* __restrict__ proj,
                        const float* __restrict__ gate, float* __restrict__ out) {
  const size_t total4 = (size_t)NB * NS * ND / 4;
  const v4f* x4 = (const v4f*)x;
  v4f* o4 = (v4f*)out;
  for (size_t i = (size_t)blockIdx.x * blockDim.x + threadIdx.x; i < total4;
       i += (size_t)gridDim.x * blockDim.x) {
    const size_t e0 = i * 4;
    const int b = (int)(e0 / ((size_t)NS * ND));
    const int d = (int)(e0 % ND);
    const float g = gate[b];
    const v4f xv = __builtin_nontemporal_load(&x4[i]);
    const v4f pv = *(const v4f*)(proj + (size_t)b * ND + d);
    v4f ov = g * pv + (1.f - g) * xv;
    __builtin_nontemporal_store(ov, &o4[i]);
  }
}

extern "C" void kernel_launch(void* const* d_in, const int* in_sizes, int n_in,
                              void* d_out, int out_size, void* d_ws, size_t ws_size,
                              hipStream_t stream) {
  (void)in_sizes; (void)n_in; (void)out_size; (void)ws_size;
  const int*   strategy = (const int*)  d_in[0];
  const float* x        = (const float*)d_in[1];
  const float* Wq  = (const float*)d_in[2];  const float* bq  = (const float*)d_in[3];
  const float* Wk  = (const float*)d_in[4];  const float* bk  = (const float*)d_in[5];
  const float* Wv  = (const float*)d_in[6];  const float* bv  = (const float*)d_in[7];
  const float* Wc  = (const float*)d_in[8];  const float* bc  = (const float*)d_in[9];
  const float* Wgin = (const float*)d_in[10]; const float* bgin = (const float*)d_in[11];
  const float* Wgout= (const float*)d_in[12];
  const float* Wfc1 = (const float*)d_in[13]; const float* bfc1 = (const float*)d_in[14];
  const float* Wfc2 = (const float*)d_in[15]; const float* bfc2 = (const float*)d_in[16];
  float* out = (float*)d_out;

  // workspace layout (floats)
  float* ws = (float*)d_ws;
  float* partial  = ws;                 // NB*PCH*ND = 524288
  float* pooled   = partial + (size_t)NB * PCH * ND;   // 16384
  float* q        = pooled  + (size_t)NB * ND;         // 16384
  float* th       = q       + (size_t)NB * ND;         // NB*NH*ND = 131072
  float* sbias    = th      + (size_t)NB * NH * ND;    // 128
  float* scores   = sbias   + NB * NH;                 // NB*NH*(NS+1) = 262272
  float* upart    = scores  + (size_t)NB * NH * (NS + 1); // JCH*NB*NH*ND = 1048576
  float* u        = upart   + (size_t)JCH * NB * NH * ND; // 131072
  float* ctx      = u       + (size_t)NB * NH * ND;    // 16384
  float* attn_out = ctx     + (size_t)NB * ND;         // 16384
  float* gate     = attn_out + (size_t)NB * ND;        // 16
  float* h1       = gate    + NB;                      // NB*FF = 65536
  float* proj     = h1      + (size_t)NB * FF;         // 16384

  const float scale = 1.0f / sqrtf((float)HD);

  k_pool_partial<<<dim3(NB * PCH, ND / 256), 256, 0, stream>>>(x, partial);
  k_pool_reduce <<<dim3(NB, ND / 256), 256, 0, stream>>>(partial, pooled);
  k_gemm16_bt   <<<8, 256, 0, stream>>>(pooled, Wq, bq, q);           // q
  k_sbias       <<<1, 128, 0, stream>>>(q, bk, sbias, scale);
  k_gemm_th     <<<64, 256, 0, stream>>>(q, Wk, th, scale);           // t = scale*q·Wk
  k_scores      <<<(NB * (NS + 1)) / 8, 256, 0, stream>>>(x, pooled, th, sbias, scores);
  k_softmax     <<<NB * NH, 256, 0, stream>>>(scores);
  k_upart       <<<dim3(ND / 256, NB, JCH), 256, 0, stream>>>(x, pooled, scores, upart);
  k_ureduce     <<<(NB * NH * ND) / 256, 256, 0, stream>>>(upart, u);
  k_gemm_ctx    <<<8, 256, 0, stream>>>(u, Wv, bv, ctx);              // ctx = Wv·u + bv
  k_gemm16_bt   <<<8, 256, 0, stream>>>(ctx, Wc, bc, attn_out);       // attn_out
  k_gate        <<<NB, 128, 0, stream>>>(attn_out, strategy, Wgin, bgin, Wgout,
                                         gate, out + (size_t)NB * NS * ND);
  k_fc1         <<<(NB * FF) / 8, 256, 0, stream>>>(attn_out, strategy, Wfc1, bfc1, h1);
  k_fc2         <<<(NB * ND) / 8, 256, 0, stream>>>(h1, strategy, Wfc2, bfc2, proj);
  k_blend       <<<4096, 256, 0, stream>>>(x, proj, gate, out);
}